// Mean_2px_Pad2d_22840636080212
// MI455X (gfx1250) — compile-verified
//
#include <hip/hip_runtime.h>

// Problem constants (from reference): x[128, 96, 64, 64] f32 -> y[128, 96, 66, 66] f32
#define C_CH      96
#define HW        64
#define OHW       66
#define PLANE_IN  (HW * HW)     // 4096 floats = 16 KB
#define PLANE_OUT (OHW * OHW)   // 4356 floats
#define BORDER_N  (2 * OHW + 2 * HW)  // 260 frame elements per plane

// ---- CDNA5 async Global<->LDS path (ASYNCcnt-tracked DMA-style transfers) ----
#if defined(__has_builtin)
#if __has_builtin(__builtin_amdgcn_global_load_async_to_lds_b128) && \
    __has_builtin(__builtin_amdgcn_global_store_async_from_lds_b32) && \
    __has_builtin(__builtin_amdgcn_s_wait_asynccnt)
#define USE_ASYNC_LDS 1
#endif
#endif
#ifndef USE_ASYNC_LDS
#define USE_ASYNC_LDS 0
#endif

// Exact pointer types per hipcc diagnostics:
//   load_async_to_lds_b128 : arg0 = global int4*  (AS1, vector_size(16))
//   store_async_from_lds_b32: arg0 = global int*   (AS1)
typedef int v4i __attribute__((vector_size(16)));
typedef __attribute__((address_space(1))) v4i* gv4i_p;
typedef __attribute__((address_space(3))) v4i* lv4i_p;
typedef __attribute__((address_space(1))) int* gi_p;
typedef __attribute__((address_space(3))) int* li_p;

__device__ __forceinline__ void decode_patch(int plane, int& py, int& px) {
  // plane = b * 96 + c ; b = image*16 + py*4 + px  (P = 4)
  const int b = plane / C_CH;
  const int p = b & 15;
  py = p >> 2;
  px = p & 3;
}

// ---------------------------------------------------------------------------
// Interior: out[plane, 1..64, 1..64] = x[plane, 0..63, 0..63]
// One block (8 wave32) per plane. HBM -> LDS -> HBM via async transfers,
// no VGPR data movement; b128 loads (16B aligned both sides), b32 stores
// (output rows are 264B-strided, so only 4B alignment is guaranteed).
// ---------------------------------------------------------------------------
__global__ __launch_bounds__(256) void pad_interior_kernel(
    const float* __restrict__ x, float* __restrict__ out) {
  const int plane = blockIdx.x;
  const int t = threadIdx.x;
  const float* xp = x + (size_t)plane * PLANE_IN;
  float* op = out + (size_t)plane * PLANE_OUT;

#if USE_ASYNC_LDS
  __shared__ float tile[PLANE_IN];  // 16 KB stage
  // Phase 1: 4 x b128 async loads per thread = full 16 KB plane into LDS.
#pragma unroll
  for (int k = 0; k < 4; ++k) {
    const int byteOff = (k * 256 + t) * 16;
    __builtin_amdgcn_global_load_async_to_lds_b128(
        (gv4i_p)((char*)const_cast<float*>(xp) + byteOff),
        (lv4i_p)((char*)tile + byteOff),
        0, 0);
  }
  __builtin_amdgcn_s_wait_asynccnt(0);  // my wave's loads done
  __syncthreads();                      // all waves' loads done
  // Phase 2: scatter to shifted interior with b32 async stores (coalesced:
  // consecutive lanes hit consecutive output columns).
#pragma unroll
  for (int k = 0; k < 16; ++k) {
    const int e = k * 256 + t;
    const int r0 = e >> 6;
    const int c = e & 63;
    __builtin_amdgcn_global_store_async_from_lds_b32(
        (gi_p)(op + (r0 + 1) * OHW + 1 + c),
        (li_p)((char*)tile + e * 4),
        0, 0);
  }
  // S_ENDPGM performs an implicit wait-idle, draining ASYNCcnt.
#else
  // Fallback: direct coalesced dword copy.
#pragma unroll
  for (int k = 0; k < 16; ++k) {
    const int e = k * 256 + t;
    const int r0 = e >> 6;
    const int c = e & 63;
    op[(r0 + 1) * OHW + 1 + c] = xp[e];
  }
#endif
}

// ---------------------------------------------------------------------------
// Border frame: 260 elements per plane (rows 0 & 65 full width, cols 0 & 65
// interior height). Mean-of-2 boundary rows/cols, edge-replicated corners,
// zeroed for patches on the image boundary (zeroing wins, applied last in ref).
// ---------------------------------------------------------------------------
__global__ void pad_border_kernel(const float* __restrict__ x,
                                  float* __restrict__ out, int nplanes) {
  const int idx = blockIdx.x * blockDim.x + threadIdx.x;
  const int total = nplanes * BORDER_N;
  if (idx >= total) return;
  const int plane = idx / BORDER_N;
  const int i = idx - plane * BORDER_N;
  int py, px;
  decode_patch(plane, py, px);
  const float* xp = x + (size_t)plane * PLANE_IN;

  int r, cc;
  if (i < OHW)               { r = 0;                    cc = i; }
  else if (i < 2 * OHW)      { r = OHW - 1;              cc = i - OHW; }
  else if (i < 2 * OHW + HW) { r = 1 + (i - 2 * OHW);    cc = 0; }
  else                       { r = 1 + (i - 2 * OHW - HW); cc = OHW - 1; }

  float v;
  if (r == 0) {
    if (py == 0)            v = 0.0f;                       // top image boundary
    else if (cc == 0)       v = (px == 0) ? 0.0f : xp[0];   // corner: edge or zero
    else if (cc == OHW - 1) v = (px == 3) ? 0.0f : xp[HW - 1];
    else                    v = 0.5f * (xp[cc - 1] + xp[HW + cc - 1]);  // rows 0,1
  } else if (r == OHW - 1) {
    if (py == 3)            v = 0.0f;                       // bottom image boundary
    else if (cc == 0)       v = (px == 0) ? 0.0f : xp[(HW - 1) * HW];
    else if (cc == OHW - 1) v = (px == 3) ? 0.0f : xp[(HW - 1) * HW + HW - 1];
    else                    v = 0.5f * (xp[(HW - 2) * HW + cc - 1] +
                                        xp[(HW - 1) * HW + cc - 1]);    // rows 62,63
  } else {
    const int rr = r - 1;
    if (cc == 0) v = (px == 0) ? 0.0f
                               : 0.5f * (xp[rr * HW] + xp[rr * HW + 1]);     // cols 0,1
    else         v = (px == 3) ? 0.0f
                               : 0.5f * (xp[rr * HW + HW - 2] + xp[rr * HW + HW - 1]);
  }
  out[(size_t)plane * PLANE_OUT + r * OHW + cc] = v;
}

extern "C" void kernel_launch(void* const* d_in, const int* in_sizes, int n_in,
                              void* d_out, int out_size, void* d_ws, size_t ws_size,
                              hipStream_t stream) {
  const float* x = (const float*)d_in[0];
  float* out = (float*)d_out;
  const int nplanes = in_sizes[0] / PLANE_IN;  // 128 * 96 = 12288

  // Bulk interior (94% of traffic): one block per plane, async HBM<->LDS.
  pad_interior_kernel<<<nplanes, 256, 0, stream>>>(x, out);

  // Frame elements.
  const int btotal = nplanes * BORDER_N;
  pad_border_kernel<<<(btotal + 255) / 256, 256, 0, stream>>>(x, out, nplanes);
}